// RandomTimeMask_27084063769080
// MI455X (gfx1250) — compile-verified
//
#include <hip/hip_runtime.h>

// RandomTimeMask: out[n,c,t] = x[n,c,t] if ((t - start[n,c]) mod L) >= L/4 else 0
// Shapes: x = [128, 12, 32768] f32, starts = [128, 12] i64, out same as x.
//
// Pure streaming kernel: 402 MB total traffic -> ~17 us at 23.3 TB/s HBM.
// Strategy: wave32, b128 non-temporal loads/stores, uniform scalar load of
// `start` per block, branchless bit-test mask (L = 2^15, mask_len = 0x2000).

typedef float v4f __attribute__((ext_vector_type(4)));

constexpr int L_LEN          = 32768;            // time dimension (power of two)
constexpr int ROW_V4         = L_LEN / 4;        // 8192 float4 per row
constexpr int BLOCKS_PER_ROW = 8;                // each block: 1024 float4 = 4096 floats
constexpr int V4_PER_BLOCK   = ROW_V4 / BLOCKS_PER_ROW; // 1024
constexpr int THREADS        = 256;              // 8 wave32 per block
constexpr int ITERS          = V4_PER_BLOCK / THREADS;  // 4 float4 per thread
// mask_len = L/4 = 8192. (d mod 2^15) >= 8192  <=>  (d & 0x6000) != 0
// (mod-2^15 keeps bits [14:0]; value >= 0x2000 iff bit13 or bit14 set)
constexpr int KEEP_BITS      = 0x6000;

__global__ __launch_bounds__(THREADS) void RandomTimeMask_kernel(
    const v4f* __restrict__ x,
    const long long* __restrict__ starts,
    v4f* __restrict__ out)
{
    // Block -> (row, segment) mapping keeps `row` wave-uniform so the
    // starts[] load is a single scalar (s_load_b64) per block.
    const int row = (int)(blockIdx.x >> 3);                 // blockIdx / BLOCKS_PER_ROW
    const int seg = (int)(blockIdx.x & (BLOCKS_PER_ROW - 1));
    const int s   = (int)starts[row];                       // 0 <= s < L, fits in i32

    const long long rowBase = (long long)row * ROW_V4;      // float4 index of row start
    const int base4 = seg * V4_PER_BLOCK + (int)threadIdx.x;

#pragma unroll
    for (int k = 0; k < ITERS; ++k) {
        const int i4 = base4 + k * THREADS;                 // float4 index within row
        const long long g = rowBase + i4;                   // global float4 index

        v4f v = __builtin_nontemporal_load(&x[g]);          // global_load_b128, TH=NT

        const int p = (i4 << 2) - s;                        // pos - start for lane .x
        v4f r;
        r.x = ((p + 0) & KEEP_BITS) ? v.x : 0.0f;
        r.y = ((p + 1) & KEEP_BITS) ? v.y : 0.0f;
        r.z = ((p + 2) & KEEP_BITS) ? v.z : 0.0f;
        r.w = ((p + 3) & KEEP_BITS) ? v.w : 0.0f;

        __builtin_nontemporal_store(r, &out[g]);            // global_store_b128, TH=NT
    }
}

extern "C" void kernel_launch(void* const* d_in, const int* in_sizes, int n_in,
                              void* d_out, int out_size, void* d_ws, size_t ws_size,
                              hipStream_t stream)
{
    const v4f*       x      = (const v4f*)d_in[0];          // [N, C, L] f32
    const long long* starts = (const long long*)d_in[1];    // [N, C]    i64
    v4f*             out    = (v4f*)d_out;

    const int rows = in_sizes[1];                           // N * C = 1536
    const int grid = rows * BLOCKS_PER_ROW;                 // 12288 blocks

    RandomTimeMask_kernel<<<grid, THREADS, 0, stream>>>(x, starts, out);
}